// Encoder_5497558139362
// MI455X (gfx1250) — compile-verified
//
#include <hip/hip_runtime.h>
#include <hip/hip_bf16.h>

typedef __attribute__((ext_vector_type(2))) float v2f;
typedef __attribute__((ext_vector_type(8))) float v8f;

// ---------------------------------------------------------------------------
// K1: tok[t] = argmax_v logits[t, v]   (first occurrence wins, like jnp.argmax)
// ---------------------------------------------------------------------------
__global__ __launch_bounds__(256)
void argmax_kernel(const float* __restrict__ logits, int* __restrict__ tok,
                   int T, int V) {
    int t = blockIdx.x * blockDim.x + threadIdx.x;
    if (t >= T) return;
    const float* p = logits + (size_t)t * V;
    float best = p[0];
    int bi = 0;
    for (int v = 1; v < V; ++v) {
        float x = p[v];
        if (x > best) { best = x; bi = v; }
    }
    tok[t] = bi;
}

// ---------------------------------------------------------------------------
// K2: single-workgroup segmentation.
//  Phase 1: change flags -> segment ids (block scan over T in 1024 chunks),
//           record seg_start[], nseg.
//  Phase 2: valid = (seg token != 0); compact scan -> row_start/row_len per
//           output row, nvalid.
// Wave32 scans via ballot+popc; cross-wave via LDS.
// ---------------------------------------------------------------------------
__global__ __launch_bounds__(1024)
void seg_scan_kernel(const int* __restrict__ tok, int* __restrict__ seg_start,
                     int* __restrict__ row_start, int* __restrict__ row_len,
                     int* __restrict__ scalars, int T) {
    __shared__ int wsum[32];
    __shared__ int s_carry;
    const int tid  = threadIdx.x;
    const int lane = tid & 31;
    const int wave = tid >> 5;

    if (tid == 0) s_carry = 0;
    __syncthreads();

    // ---- Phase 1: segment ids / boundaries ----
    for (int base = 0; base < T; base += 1024) {
        int t    = base + tid;
        int my   = tok[t];
        int prev = (t == 0) ? -1 : tok[t - 1];
        int change = (my != prev) ? 1 : 0;

        unsigned m  = (unsigned)__ballot(change);
        int incl    = __popc(m & (0xFFFFFFFFu >> (31 - lane)));
        if (lane == 31) wsum[wave] = incl;
        __syncthreads();
        if (wave == 0) {
            int v = wsum[lane];
            #pragma unroll
            for (int d = 1; d < 32; d <<= 1) {
                int u = __shfl_up(v, d);
                if (lane >= d) v += u;
            }
            wsum[lane] = v;   // inclusive wave-sum scan
        }
        __syncthreads();
        int carry      = s_carry;
        int excl_waves = (wave == 0) ? 0 : wsum[wave - 1];
        int seg        = carry + excl_waves + incl - 1;
        if (change) seg_start[seg] = t;
        __syncthreads();
        if (tid == 1023) s_carry = carry + excl_waves + incl;  // chunk total
        __syncthreads();
    }

    int nseg = s_carry;
    if (tid == 0) {
        scalars[0]      = nseg;
        seg_start[nseg] = T;   // sentinel
        s_carry         = 0;
    }
    __threadfence();
    __syncthreads();

    // ---- Phase 2: compact valid (non-blank) segments ----
    for (int base = 0; base < nseg; base += 1024) {
        int s = base + tid;
        int valid = 0, st = 0, len = 0;
        if (s < nseg) {
            st  = seg_start[s];
            len = seg_start[s + 1] - st;
            valid = (tok[st] != 0) ? 1 : 0;
        }
        unsigned m = (unsigned)__ballot(valid);
        int incl   = __popc(m & (0xFFFFFFFFu >> (31 - lane)));
        if (lane == 31) wsum[wave] = incl;
        __syncthreads();
        if (wave == 0) {
            int v = wsum[lane];
            #pragma unroll
            for (int d = 1; d < 32; d <<= 1) {
                int u = __shfl_up(v, d);
                if (lane >= d) v += u;
            }
            wsum[lane] = v;
        }
        __syncthreads();
        int carry      = s_carry;
        int excl_waves = (wave == 0) ? 0 : wsum[wave - 1];
        int dst        = carry + excl_waves + incl - 1;
        if (valid) { row_start[dst] = st; row_len[dst] = len; }
        __syncthreads();
        if (tid == 1023) s_carry = carry + excl_waves + incl;
        __syncthreads();
    }
    if (tid == 0) scalars[1] = s_carry;  // nvalid
}

// ---------------------------------------------------------------------------
// K3: out[r, d] = (r < nvalid) ? mean(hidden[rows of run r]) . W[d,:] + b[d] : 0
// fp32 WMMA 16x16x4 GEMM. Block = 256 threads (8 waves), tile 128(M) x 256(N),
// KB=32 k-chunk staged in LDS. Segment mean fused into A-tile load.
// Wave grid 4(M) x 2(N): each wave owns 32x128 = 2x8 tiles of 16x16 (128 acc VGPRs).
// ---------------------------------------------------------------------------
#define KB  32
#define LDP (KB + 2)   // 34 dwords: even (8B-aligned float2) + conflict padding

__global__ __launch_bounds__(256)
void gemm_kernel(const float* __restrict__ hidden, const float* __restrict__ W,
                 const float* __restrict__ bias, float* __restrict__ out,
                 const int* __restrict__ row_start, const int* __restrict__ row_len,
                 const int* __restrict__ scalars, int H, int D) {
    const int nvalid = scalars[1];
    const int R0  = blockIdx.x * 128;
    const int N0  = blockIdx.y * 256;
    const int tid = threadIdx.x;

    if (R0 >= nvalid) {
        // Dead tile: pure zero-fill (reference masks these rows to exactly 0).
        float4 z = {0.f, 0.f, 0.f, 0.f};
        for (int i = tid; i < 128 * 64; i += 256) {
            int r = i >> 6, c4 = i & 63;
            *(float4*)(out + (size_t)(R0 + r) * D + N0 + c4 * 4) = z;
        }
        return;
    }

    __shared__ float Alds[128][LDP];
    __shared__ float Blds[256][LDP];

    const int lane = tid & 31;
    const int wave = tid >> 5;
    const int wm   = wave >> 1;     // 0..3  (M position)
    const int wn   = wave & 1;      // 0..1  (N position)
    const int half = lane >> 4;     // lane half selects K pair per ISA layout
    const int mrow = lane & 15;

    v8f acc[2][8];
    #pragma unroll
    for (int i = 0; i < 2; ++i)
        #pragma unroll
        for (int j = 0; j < 8; ++j)
            #pragma unroll
            for (int q = 0; q < 8; ++q) acc[i][j][q] = 0.f;

    // A-tile loader mapping: 8 threads/row * 4 floats, 4 rows per thread.
    const int ar = tid >> 3;            // base local row 0..31 (stride 32 -> 128 rows)
    const int ak = (tid & 7) * 4;       // k sub-offset (float4)
    int   srcs[4]; int lens[4]; float inv[4];
    #pragma unroll
    for (int rr = 0; rr < 4; ++rr) {
        int rg = R0 + ar + rr * 32;
        if (rg < nvalid) {
            srcs[rr] = row_start[rg];
            lens[rr] = row_len[rg];
            inv[rr]  = 1.0f / (float)lens[rr];
        } else {
            srcs[rr] = 0; lens[rr] = 0; inv[rr] = 0.f;  // zero row
        }
    }

    for (int k0 = 0; k0 < H; k0 += KB) {
        // ---- stage A tile: fused segment mean ----
        #pragma unroll
        for (int rr = 0; rr < 4; ++rr) {
            int r = ar + rr * 32;
            float sx = 0.f, sy = 0.f, sz = 0.f, sw = 0.f;
            const float* hp = hidden + (size_t)srcs[rr] * H + k0 + ak;
            for (int i = 0; i < lens[rr]; ++i) {
                float4 x = *(const float4*)hp;
                sx += x.x; sy += x.y; sz += x.z; sw += x.w;
                hp += H;
            }
            Alds[r][ak + 0] = sx * inv[rr];
            Alds[r][ak + 1] = sy * inv[rr];
            Alds[r][ak + 2] = sz * inv[rr];
            Alds[r][ak + 3] = sw * inv[rr];
        }
        // ---- stage B tile from W (rows = output cols, k-contiguous) ----
        #pragma unroll
        for (int rr = 0; rr < 8; ++rr) {
            int d = ar + rr * 32;
            float4 x = *(const float4*)(W + (size_t)(N0 + d) * H + k0 + ak);
            Blds[d][ak + 0] = x.x;
            Blds[d][ak + 1] = x.y;
            Blds[d][ak + 2] = x.z;
            Blds[d][ak + 3] = x.w;
        }
        if (k0 + KB < H) {  // pull next chunk toward caches (global_prefetch_b8)
            __builtin_prefetch(hidden + (size_t)srcs[0] * H + k0 + KB + ak, 0, 1);
            __builtin_prefetch(W + (size_t)(N0 + ar) * H + k0 + KB + ak, 0, 1);
        }
        __syncthreads();

        // ---- 8 k-steps of 16 WMMAs (f32 16x16x4) ----
        #pragma unroll
        for (int ks = 0; ks < KB; ks += 4) {
            v2f afr[2], bfr[8];
            #pragma unroll
            for (int mt = 0; mt < 2; ++mt) {
                int r = wm * 32 + mt * 16 + mrow;
                afr[mt] = *(const v2f*)&Alds[r][ks + 2 * half];
            }
            #pragma unroll
            for (int nt = 0; nt < 8; ++nt) {
                int n = wn * 128 + nt * 16 + mrow;
                bfr[nt] = *(const v2f*)&Blds[n][ks + 2 * half];
            }
            #pragma unroll
            for (int mt = 0; mt < 2; ++mt)
                #pragma unroll
                for (int nt = 0; nt < 8; ++nt)
                    acc[mt][nt] = __builtin_amdgcn_wmma_f32_16x16x4_f32(
                        false, afr[mt], false, bfr[nt],
                        (short)0, acc[mt][nt], false, false);
        }
        __syncthreads();
    }

    // ---- store: C/D layout => VGPR j holds (M = j + 8*half, N = lane&15) ----
    #pragma unroll
    for (int mt = 0; mt < 2; ++mt) {
        #pragma unroll
        for (int nt = 0; nt < 8; ++nt) {
            int col  = N0 + wn * 128 + nt * 16 + mrow;
            float bv = bias[col];
            #pragma unroll
            for (int j = 0; j < 8; ++j) {
                int row = R0 + wm * 32 + mt * 16 + half * 8 + j;
                float v = (row < nvalid) ? (acc[mt][nt][j] + bv) : 0.f;
                out[(size_t)row * D + col] = v;
            }
        }
    }
}

// ---------------------------------------------------------------------------
// Host entry
// ---------------------------------------------------------------------------
extern "C" void kernel_launch(void* const* d_in, const int* in_sizes, int n_in,
                              void* d_out, int out_size, void* d_ws, size_t ws_size,
                              hipStream_t stream) {
    const float* hidden = (const float*)d_in[0];  // [1,T,H]
    const float* logits = (const float*)d_in[1];  // [1,T,V]
    const float* W      = (const float*)d_in[2];  // [D,H]
    const float* b      = (const float*)d_in[3];  // [D]

    const int D = in_sizes[3];
    const int H = in_sizes[2] / D;
    const int T = in_sizes[0] / H;
    const int V = in_sizes[1] / T;

    // workspace layout (ints): scalars[4] | tok[T] | seg_start[T+1] | row_start[T] | row_len[T]
    int* scalars   = (int*)d_ws;
    int* tok       = scalars + 4;
    int* seg_start = tok + T;
    int* row_start = seg_start + T + 1;
    int* row_len   = row_start + T;

    float* out = (float*)d_out;

    argmax_kernel<<<(T + 255) / 256, 256, 0, stream>>>(logits, tok, T, V);
    seg_scan_kernel<<<1, 1024, 0, stream>>>(tok, seg_start, row_start, row_len,
                                            scalars, T);
    dim3 grid(T / 128, D / 256);
    gemm_kernel<<<grid, 256, 0, stream>>>(hidden, W, b, out,
                                          row_start, row_len, scalars, H, D);
}